// CaptionDetectionTargetLayer_57423712747864
// MI455X (gfx1250) — compile-verified
//
#include <hip/hip_runtime.h>
#include <cstdint>
#include <math.h>

#define B_IMG   8
#define N_PROP  4096
#define N_GT    512
#define T_ROIS  200
#define POS_MAX 66
#define CAP_LEN 15

// -------- LDS byte-offset of a __shared__ object (low 32 bits of the flat
// aperture address are the wave-relative LDS offset on gfx1250) --------
static __device__ __forceinline__ uint32_t lds_off(const void* p) {
    return (uint32_t)(uintptr_t)p;
}

// ===================================================================
// Kernel A: per-proposal max-IoU + argmax over 512 gt boxes.
// Grid: 128 blocks (16 per image) x 256 threads, 1 proposal per thread.
// gt boxes staged to LDS via CDNA5 async global->LDS DMA (ASYNCcnt).
// Writes packed word per proposal: assign[15:0] | flag[17:16]
//   flag: 0 = invalid proposal, 1 = positive, 2 = negative
// ===================================================================
__global__ __launch_bounds__(256) void iou_argmax_kernel(
    const float* __restrict__ proposals,   // [B, 4096, 4]
    const float* __restrict__ gt_boxes,    // [B, 512, 4]
    int* __restrict__ ws)                  // [B, 4096] packed
{
    __shared__ float4 s_gt[N_GT];          // 8 KB
    __shared__ float  s_ga2[N_GT];         // area, <0 encodes invalid gt

    const int tid  = threadIdx.x;
    const int b    = blockIdx.x >> 4;
    const int tile = blockIdx.x & 15;

    // ---- async DMA: 512 x float4 (8KB) of gt boxes into LDS ----
    const float4* gt4 = reinterpret_cast<const float4*>(gt_boxes + (size_t)b * N_GT * 4);
    {
        const float4* ga0 = gt4 + tid;
        const float4* ga1 = gt4 + tid + 256;
        uint32_t l0 = lds_off(&s_gt[tid]);
        uint32_t l1 = lds_off(&s_gt[tid + 256]);
        asm volatile("global_load_async_to_lds_b128 %0, %1, off"
                     :: "v"(l0), "v"(ga0) : "memory");
        asm volatile("global_load_async_to_lds_b128 %0, %1, off"
                     :: "v"(l1), "v"(ga1) : "memory");
        asm volatile("s_wait_asynccnt 0x0" ::: "memory");
    }
    __syncthreads();

    // precompute gt areas + validity (validity folded into sign of area)
    #pragma unroll
    for (int i = 0; i < 2; ++i) {
        int g = tid + i * 256;
        float4 gb = s_gt[g];
        float sabs = fabsf(gb.x) + fabsf(gb.y) + fabsf(gb.z) + fabsf(gb.w);
        float a2 = (gb.z - gb.x) * (gb.w - gb.y);
        s_ga2[g] = (sabs > 0.0f) ? a2 : -1.0f;
    }
    __syncthreads();

    const int p = tile * 256 + tid;
    const float4* prop4 = reinterpret_cast<const float4*>(proposals + (size_t)b * N_PROP * 4);
    float4 box = prop4[p];
    float vsum = fabsf(box.x) + fabsf(box.y) + fabsf(box.z) + fabsf(box.w);
    bool  validp = vsum > 0.0f;
    float a1 = (box.z - box.x) * (box.w - box.y);

    // running best as (inter, union') pair, compared by cross-multiplication
    // (unions are positive; avoids 512 divides). init pair (-2,1) < (-1,1).
    float bi = -2.0f, bu = 1.0f;
    int   bidx = 0;
    for (int g = 0; g < N_GT; ++g) {
        float4 gb = s_gt[g];
        float a2 = s_ga2[g];
        float y1 = fmaxf(box.x, gb.x);
        float x1 = fmaxf(box.y, gb.y);
        float y2 = fminf(box.z, gb.z);
        float x2 = fminf(box.w, gb.w);
        float inter = fmaxf(x2 - x1, 0.0f) * fmaxf(y2 - y1, 0.0f);
        float uni = a1 + a2 - inter;
        float up  = (uni > 0.0f) ? uni : 1.0f;
        bool  gvalid = (a2 >= 0.0f);
        float ci = gvalid ? inter : -1.0f;     // overlap = -1 for invalid gt
        float cu = gvalid ? up    :  1.0f;
        // first-occurrence argmax: strict improvement only
        bool better = (ci * bu) > (bi * cu);
        if (better) { bi = ci; bu = cu; bidx = g; }
    }
    float maxiou = bi / bu;                    // one divide per proposal
    int flag = validp ? ((maxiou >= 0.5f) ? 1 : 2) : 0;
    ws[(size_t)b * N_PROP + p] = (bidx & 0xFFFF) | (flag << 16);
}

// ===================================================================
// Kernel B: ordered selection (wave32 ballot prefix-compaction, matching
// jnp.where(..., size=k) first-index semantics) + output assembly.
// Grid: 8 blocks x 256 threads (block == image).
// ===================================================================
__global__ __launch_bounds__(256) void select_output_kernel(
    const float* __restrict__ proposals,
    const float* __restrict__ gt_boxes,
    const int*   __restrict__ gt_caps,     // [B, 512, 15]
    const float* __restrict__ scores,      // [B, 512]
    const int*   __restrict__ ws,          // [B, 4096] packed
    float* __restrict__ out)               // rois | deltas | caps | scores
{
    __shared__ int s_packed[N_PROP];       // 16 KB
    __shared__ int s_posidx[POS_MAX];
    __shared__ int s_negidx[T_ROIS];
    __shared__ int s_cnt[2];

    const int b   = blockIdx.x;
    const int tid = threadIdx.x;

    // stage packed flags/assign to LDS (coalesced)
    for (int i = tid; i < N_PROP; i += 256)
        s_packed[i] = ws[(size_t)b * N_PROP + i];
    __syncthreads();

    // ---- wave 0: ordered compaction via ballot/popc ----
    if (tid < 32) {
        const int lane = tid;
        const unsigned lt = (1u << lane) - 1u;
        int pos_base = 0, neg_base = 0;
        for (int it = 0; it < N_PROP / 32; ++it) {
            int p = it * 32 + lane;
            int f = (s_packed[p] >> 16) & 3;
            unsigned bp = __builtin_amdgcn_ballot_w32(f == 1);
            unsigned bn = __builtin_amdgcn_ballot_w32(f == 2);
            if (f == 1) {
                int idx = pos_base + __popc(bp & lt);
                if (idx < POS_MAX) s_posidx[idx] = p;
            }
            if (f == 2) {
                int idx = neg_base + __popc(bn & lt);
                if (idx < T_ROIS) s_negidx[idx] = p;
            }
            pos_base += __popc(bp);
            neg_base += __popc(bn);
        }
        if (lane == 0) {
            int pos_cnt = pos_base < POS_MAX ? pos_base : POS_MAX;
            // neg_target = int(pos_cnt/0.33f) - pos_cnt  (f32 div, trunc)
            int neg_target = (int)((float)pos_cnt / 0.33f) - pos_cnt;
            int neg_cnt = neg_target;
            if (neg_base < neg_cnt)        neg_cnt = neg_base;
            if (T_ROIS - pos_cnt < neg_cnt) neg_cnt = T_ROIS - pos_cnt;
            if (neg_cnt < 0)               neg_cnt = 0;
            s_cnt[0] = pos_cnt;
            s_cnt[1] = neg_cnt;
        }
    }
    __syncthreads();

    // ---- output assembly: one thread per ROI slot ----
    const int pos_cnt = s_cnt[0];
    const int neg_cnt = s_cnt[1];

    float* out_rois   = out;
    float* out_deltas = out + (size_t)B_IMG * T_ROIS * 4;
    float* out_caps   = out + (size_t)B_IMG * T_ROIS * 8;
    float* out_scores = out + (size_t)B_IMG * T_ROIS * 8 + (size_t)B_IMG * T_ROIS * CAP_LEN;

    if (tid < T_ROIS) {
        const int t = tid;
        const float4* prop4 = reinterpret_cast<const float4*>(proposals + (size_t)b * N_PROP * 4);
        const float4* gt4   = reinterpret_cast<const float4*>(gt_boxes  + (size_t)b * N_GT * 4);

        float r0=0.f,r1=0.f,r2=0.f,r3=0.f;
        float d0=0.f,d1=0.f,d2=0.f,d3=0.f;
        float sc=0.f;
        float caps[CAP_LEN];
        #pragma unroll
        for (int j = 0; j < CAP_LEN; ++j) caps[j] = 0.f;

        if (t < pos_cnt) {
            int p = s_posidx[t];
            int g = s_packed[p] & 0xFFFF;
            float4 box = prop4[p];
            float4 gb  = gt4[g];
            r0 = box.x; r1 = box.y; r2 = box.z; r3 = box.w;
            float h  = box.z - box.x, w  = box.w - box.y;
            float cy = box.x + 0.5f * h, cx = box.y + 0.5f * w;
            float gh = gb.z - gb.x, gw = gb.w - gb.y;
            float gcy = gb.x + 0.5f * gh, gcx = gb.y + 0.5f * gw;
            d0 = ((gcy - cy) / h) / 0.1f;
            d1 = ((gcx - cx) / w) / 0.1f;
            d2 = logf(gh / h) / 0.2f;
            d3 = logf(gw / w) / 0.2f;
            const int* cp = gt_caps + ((size_t)b * N_GT + g) * CAP_LEN;
            #pragma unroll
            for (int j = 0; j < CAP_LEN; ++j) caps[j] = (float)cp[j];
            sc = scores[(size_t)b * N_GT + g];
        } else if (t < pos_cnt + neg_cnt) {
            int p = s_negidx[t - pos_cnt];
            float4 box = prop4[p];
            r0 = box.x; r1 = box.y; r2 = box.z; r3 = box.w;
        }

        size_t rb = ((size_t)b * T_ROIS + t) * 4;
        out_rois[rb + 0] = r0; out_rois[rb + 1] = r1;
        out_rois[rb + 2] = r2; out_rois[rb + 3] = r3;
        out_deltas[rb + 0] = d0; out_deltas[rb + 1] = d1;
        out_deltas[rb + 2] = d2; out_deltas[rb + 3] = d3;
        size_t cb = ((size_t)b * T_ROIS + t) * CAP_LEN;
        #pragma unroll
        for (int j = 0; j < CAP_LEN; ++j) out_caps[cb + j] = caps[j];
        out_scores[(size_t)b * T_ROIS + t] = sc;
    }
}

extern "C" void kernel_launch(void* const* d_in, const int* in_sizes, int n_in,
                              void* d_out, int out_size, void* d_ws, size_t ws_size,
                              hipStream_t stream) {
    const float* proposals = (const float*)d_in[0];   // [8,4096,4] f32
    const float* gt_boxes  = (const float*)d_in[1];   // [8,512,4]  f32
    const int*   gt_caps   = (const int*)  d_in[2];   // [8,512,15] i32
    const float* scores    = (const float*)d_in[3];   // [8,512]    f32
    float* out = (float*)d_out;                       // 38400 f32
    int*   ws  = (int*)d_ws;                          // needs 128 KB

    iou_argmax_kernel<<<dim3(B_IMG * 16), dim3(256), 0, stream>>>(proposals, gt_boxes, ws);
    select_output_kernel<<<dim3(B_IMG), dim3(256), 0, stream>>>(proposals, gt_boxes, gt_caps,
                                                                scores, ws, out);
}